// LoRAPool_47227460386885
// MI455X (gfx1250) — compile-verified
//
#include <hip/hip_runtime.h>

// ---------------------------------------------------------------------------
// LoRA-MoE pool: out[t,d] = sum_{e,r} gate[t,e]*2.0 * (h[t,:]·A[e,r,:]) * B[e,d,r]
// h: [16384, 2048] f32, p_L: [16384, 8] f32, A: [8,16,2048] f32, B: [8,2048,16] f32
// Memory-bound (256 MB @ 23.3 TB/s ≈ 11 us). Strategy: stream h once through
// LDS (double-buffered async global->LDS loads), fp32 WMMA (16x16x4) for both
// projections, (e,r) folded into K=128 for the up-projection. ws holds
// pre-swizzled operand layouts of A and B.
// ---------------------------------------------------------------------------

#define D_MODEL   2048
#define N_EXPERTS 8
#define RANK      16
#define KUP       (N_EXPERTS * RANK)   // 128
#define SCALING   2.0f
#define CHUNK     256                  // f32 columns of h staged per buffer
#define CS        (CHUNK + 4)          // padded LDS row stride -> conflict free (260%64==4)
#define HS        (KUP + 4)            // padded LDS stride for HRg (132%64==4)
#define NCHUNK    (D_MODEL / CHUNK)    // 8
#define LPT       8                    // b128 loads per thread per chunk (16*256/4/128)

#if defined(__has_builtin)
#if __has_builtin(__builtin_amdgcn_global_load_async_to_lds_b128) && \
    __has_builtin(__builtin_amdgcn_s_wait_asynccnt)
#define USE_ASYNC_LDS 1
#endif
#endif

typedef float v2f __attribute__((ext_vector_type(2)));
typedef float v8f __attribute__((ext_vector_type(8)));
typedef int   v4i_vs __attribute__((vector_size(4 * sizeof(int))));
typedef __attribute__((address_space(1))) v4i_vs* gas_v4i_ptr;   // global
typedef __attribute__((address_space(3))) v4i_vs* las_v4i_ptr;   // LDS

static __device__ __forceinline__ v8f wmma4(v2f a, v2f b, v8f c) {
    // V_WMMA_F32_16X16X4_F32 : D = A(16x4) * B(4x16) + C(16x16), fp32 exact
    return __builtin_amdgcn_wmma_f32_16x16x4_f32(
        /*neg_a=*/false, a, /*neg_b=*/false, b,
        /*c_mod=*/(short)0, c, /*reuse_a=*/false, /*reuse_b=*/false);
}

// ---------------------------------------------------------------------------
// Prep: swizzle A and B into WMMA-operand-friendly, coalescable layouts.
//   Ap[e][(d>>1)][r][d&1]        (element (e,r,d)) -> lane r float2 = A[e,r,d..d+1]
//   Bp[(k>>1)][d][k&1], k=e*16+r (element (k,d))   -> lane d float2 = Bf[k..k+1, d]
// ---------------------------------------------------------------------------
__global__ void lora_prep_kernel(const float* __restrict__ A,
                                 const float* __restrict__ B,
                                 float* __restrict__ Ap,
                                 float* __restrict__ Bp) {
    int idx = blockIdx.x * blockDim.x + threadIdx.x;
    if (idx >= N_EXPERTS * RANK * D_MODEL) return;
    int d = idx % D_MODEL;
    int t = idx / D_MODEL;
    int r = t % RANK;
    int e = t / RANK;
    // A source layout [E][R][D]
    Ap[(size_t)e * (D_MODEL * RANK) + (size_t)(d >> 1) * (RANK * 2) + r * 2 + (d & 1)] = A[idx];
    // B source layout [E][D][R]
    float bv = B[((size_t)e * D_MODEL + d) * RANK + r];
    int k = e * RANK + r;
    Bp[((size_t)(k >> 1) * D_MODEL + d) * 2 + (k & 1)] = bv;
}

// ---------------------------------------------------------------------------
// Main kernel: one 128-thread block (4 wave32) per 16-token tile.
// ---------------------------------------------------------------------------
__global__ __launch_bounds__(128)
void lora_moe_kernel(const float* __restrict__ h,
                     const float* __restrict__ pL,
                     const float* __restrict__ Ap,
                     const float* __restrict__ Bp,
                     float* __restrict__ out) {
    __shared__ float hs[2][16 * CS];       // double-buffered h chunks (2 x 16.6 KB)
    __shared__ float hrs[16 * HS];         // gated HRg[16x128] (8.4 KB)
    __shared__ float gl[16 * N_EXPERTS];   // gates * SCALING

    const int tok0 = blockIdx.x * 16;
    const int tid  = threadIdx.x;
    const int wave = tid >> 5;
    const int lane = tid & 31;
    const int m    = lane & 15;            // row (A-op) / col (B-op) index
    const int co   = (lane >> 4) << 1;     // K sub-offset: 0 (lanes 0-15) or 2 (16-31)
    const int hi8  = (lane >> 4) << 3;     // C/D row base: 0 or 8

    // ---- Phase 0: top-2 gates for the 16 tokens of this tile ----
    if (tid < 16) {
        float p[N_EXPERTS];
        #pragma unroll
        for (int e = 0; e < N_EXPERTS; ++e) p[e] = pL[(size_t)(tok0 + tid) * N_EXPERTS + e];
        int i1 = 0;
        #pragma unroll
        for (int e = 1; e < N_EXPERTS; ++e) if (p[e] > p[i1]) i1 = e;
        int i2 = (i1 == 0) ? 1 : 0;
        #pragma unroll
        for (int e = 0; e < N_EXPERTS; ++e) if (e != i1 && p[e] > p[i2]) i2 = e;
        #pragma unroll
        for (int e = 0; e < N_EXPERTS; ++e)
            gl[tid * N_EXPERTS + e] = (e == i1 || e == i2) ? SCALING * p[e] : 0.0f;
    }

    // Chunk stage: 16 rows x 256 floats, 8 x b128 per thread, coalesced.
    auto stage_chunk = [&](int buf, int kc) {
        #pragma unroll
        for (int j = 0; j < LPT; ++j) {
            int f   = j * 128 + tid;           // float4 index within tile
            int row = f >> 6;                  // 64 float4 per row
            int c4  = (f & 63) << 2;           // float column
            const float* src = h + (size_t)(tok0 + row) * D_MODEL + kc + c4;
            float*       dst = &hs[buf][row * CS + c4];
#if USE_ASYNC_LDS
            __builtin_amdgcn_global_load_async_to_lds_b128(
                (gas_v4i_ptr)src, (las_v4i_ptr)dst, 0, 0);
#else
            *(float4*)dst = *(const float4*)src;
#endif
        }
    };

    // ---- Phase 1: down-projection. Each wave owns 2 experts. ----
    const int e0 = wave * 2;
    const int e1 = e0 + 1;
    v8f acc0 = {};
    v8f acc1 = {};

    stage_chunk(0, 0);                      // prime the pipeline
    int cur = 0;
    for (int i = 0; i < NCHUNK; ++i) {
        const int kc = i * CHUNK;
        if (i + 1 < NCHUNK) {
            stage_chunk(cur ^ 1, kc + CHUNK);   // nxt buffer is free (barrier below ensured)
#if USE_ASYNC_LDS
            __builtin_amdgcn_s_wait_asynccnt(LPT);  // chunk i retired; chunk i+1 in flight
#endif
        } else {
#if USE_ASYNC_LDS
            __builtin_amdgcn_s_wait_asynccnt(0);    // drain final chunk
#endif
        }
        __syncthreads();                    // chunk i visible to all waves

        #pragma unroll 4
        for (int ks = 0; ks < CHUNK; ks += 4) {
            // A operand: h rows (M = m), K pair at ks+co
            v2f af = *(const v2f*)(&hs[cur][m * CS + ks + co]);
            // B operand: LoRA-A columns (N = m = r), K pair from pre-swizzled Ap
            int p2 = (kc + ks + co) >> 1;
            v2f b0 = *(const v2f*)(Ap + (size_t)e0 * (D_MODEL * RANK) + (size_t)p2 * (RANK * 2) + m * 2);
            v2f b1 = *(const v2f*)(Ap + (size_t)e1 * (D_MODEL * RANK) + (size_t)p2 * (RANK * 2) + m * 2);
            acc0 = wmma4(af, b0, acc0);
            acc1 = wmma4(af, b1, acc1);
        }
        __syncthreads();                    // all waves done reading hs[cur] -> it is free
        cur ^= 1;
    }

    // ---- Phase 2: apply gates, lay HRg down in LDS as [token][k=e*16+r] ----
    #pragma unroll
    for (int v = 0; v < 8; ++v) {
        int row = v + hi8;                 // token row (C/D layout: M = v + 8*(lane>=16))
        float g0 = gl[row * N_EXPERTS + e0];
        float g1 = gl[row * N_EXPERTS + e1];
        hrs[row * HS + e0 * RANK + m] = acc0[v] * g0;
        hrs[row * HS + e1 * RANK + m] = acc1[v] * g1;
    }
    __syncthreads();

    // ---- Phase 3: up-projection out[16 x 2048] = HRg[16x128] x Bf[128x2048] ----
    const int d0base = wave * (D_MODEL / 4);        // 512 columns per wave
    for (int nt = 0; nt < (D_MODEL / 4) / 16; ++nt) {
        const int d0 = d0base + nt * 16;
        v8f acc = {};
        #pragma unroll
        for (int k0 = 0; k0 < KUP; k0 += 4) {
            v2f af = *(const v2f*)(&hrs[m * HS + k0 + co]);          // A op from LDS
            int p2 = (k0 + co) >> 1;
            v2f bf = *(const v2f*)(Bp + ((size_t)p2 * D_MODEL + d0 + m) * 2);  // B op, coalesced
            acc = wmma4(af, bf, acc);
        }
        #pragma unroll
        for (int v = 0; v < 8; ++v) {
            int row = v + hi8;
            out[(size_t)(tok0 + row) * D_MODEL + d0 + m] = acc[v];
        }
    }
}

// ---------------------------------------------------------------------------
extern "C" void kernel_launch(void* const* d_in, const int* in_sizes, int n_in,
                              void* d_out, int out_size, void* d_ws, size_t ws_size,
                              hipStream_t stream) {
    (void)in_sizes; (void)n_in; (void)out_size; (void)ws_size;
    const float* h  = (const float*)d_in[0];
    const float* pL = (const float*)d_in[1];
    const float* A  = (const float*)d_in[2];
    const float* B  = (const float*)d_in[3];
    float* out = (float*)d_out;

    // ws: Ap (1 MB) | Bp (1 MB)
    float* Ap = (float*)d_ws;
    float* Bp = Ap + (size_t)N_EXPERTS * RANK * D_MODEL;

    const int prep_n = N_EXPERTS * RANK * D_MODEL;   // 262144
    lora_prep_kernel<<<(prep_n + 255) / 256, 256, 0, stream>>>(A, B, Ap, Bp);

    const int n_tiles = (4 * 4096) / 16;             // 1024 token tiles
    lora_moe_kernel<<<n_tiles, 128, 0, stream>>>(h, pL, Ap, Bp, out);
}